// Classifier_52166672777735
// MI455X (gfx1250) — compile-verified
//
#include <hip/hip_runtime.h>
#include <math.h>

#define DD 1024
#define NA 1887
#define BB 8
#define TT 16
#define ROWS 128   // B*T

typedef __attribute__((ext_vector_type(2))) float v2f;
typedef __attribute__((ext_vector_type(8))) float v8f;

// ---------------------------------------------------------------------------
// Pooling: x[B,T,D,7,7,3] -> mean over HW.
//   c==0 -> A0 [T,B,D]  (action stream, pre-transposed)
//   c==1 -> F1 [B,T,D]  (noun stream)
//   c==2 -> F2 [B,T,D]  (verb stream)
// ---------------------------------------------------------------------------
__global__ void pool_kernel(const float* __restrict__ x, float* __restrict__ A0,
                            float* __restrict__ F1, float* __restrict__ F2) {
  int tid = blockIdx.x * blockDim.x + threadIdx.x;
  if (tid >= ROWS * DD * 3) return;
  int c  = tid % 3;
  int rd = tid / 3;          // r*D + d, r = b*T + t
  int d  = rd % DD;
  int r  = rd / DD;
  const float* p = x + (size_t)rd * 147 + c;   // 147 = 7*7*3
  float s = 0.f;
  #pragma unroll
  for (int i = 0; i < 49; ++i) s += p[i * 3];
  s *= (1.0f / 49.0f);
  if (c == 0) {
    int b = r >> 4, t = r & 15;                // r = b*16 + t
    A0[(size_t)(t * BB + b) * DD + d] = s;     // [T,B] order
  } else if (c == 1) {
    F1[(size_t)r * DD + d] = s;
  } else {
    F2[(size_t)r * DD + d] = s;
  }
}

// ---------------------------------------------------------------------------
// f32 WMMA K-loop over a 16x16 output tile. A-frag: 16x4 (lane m=l&15,
// koff=(l>>4)*2 holds K pair per documented layout). B = W^T tile (lane n).
// ---------------------------------------------------------------------------
__device__ __forceinline__ v8f wmma_k_loop(const float* __restrict__ arow,
                                           const float* __restrict__ wrow,
                                           int K, v8f acc) {
  #pragma unroll 4
  for (int k = 0; k < K; k += 4) {
    v2f a = *(const v2f*)(arow + k);
    v2f b = *(const v2f*)(wrow + k);
    acc = __builtin_amdgcn_wmma_f32_16x16x4_f32(false, a, false, b,
                                                (short)0, acc, false, false);
  }
  return acc;
}

// MODE 0: out[r*ldo + coloff + n] = acc            (plain, + up to 2 biases)
// MODE 1: fuse store: row remap b*T+t -> t*B+b, cols<D -> out(h0), else out2(c0)
// MODE 2: final: row remap t*B+b -> b*T+t, guard n<N (N=1887)
template <int MODE>
__global__ void gemm_kernel(const float* __restrict__ A, int lda,
                            const float* __restrict__ W, int ldw,
                            const float* __restrict__ bias1,
                            const float* __restrict__ bias2,
                            float* __restrict__ out, int ldo, int coloff,
                            float* __restrict__ out2, int N, int K) {
  int wave = (blockIdx.x * blockDim.x + threadIdx.x) >> 5;
  int lane = threadIdx.x & 31;
  int m0 = (wave & 7) << 4;          // 128 rows -> 8 row tiles
  int n0 = (wave >> 3) << 4;
  int mr = lane & 15;
  int koff = (lane >> 4) << 1;
  int hi = lane >> 4;
  int ncol = n0 + (lane & 15);
  int nload = (ncol < N) ? ncol : (N - 1);   // clamp W row for ragged N
  const float* arow = A + (size_t)(m0 + mr) * lda + koff;
  const float* wrow = W + (size_t)nload * ldw + koff;
  float bsum = bias1[nload] + (bias2 ? bias2[nload] : 0.f);
  v8f acc;
  #pragma unroll
  for (int i = 0; i < 8; ++i) acc[i] = bsum;
  acc = wmma_k_loop(arow, wrow, K, acc);
  #pragma unroll
  for (int v = 0; v < 8; ++v) {
    int r = m0 + v + (hi << 3);      // documented 16x16 f32 C/D layout
    float val = acc[v];
    if (MODE == 0) {
      out[(size_t)r * ldo + coloff + ncol] = val;
    } else if (MODE == 1) {
      int b = r >> 4, t = r & 15;    // r = b*T+t
      int rr = t * BB + b;           // -> [T,B]
      if (ncol < DD) out[(size_t)rr * DD + ncol] = val;
      else           out2[(size_t)rr * DD + (ncol - DD)] = val;
    } else {
      if (ncol < N) {
        int t = r >> 3, b = r & 7;   // r = t*B+b
        out[(size_t)(b * TT + t) * N + ncol] = val;   // [B,T,NA]
      }
    }
  }
}

// ---------------------------------------------------------------------------
// One LSTM step, fused, with dead-row pruning and split-K=2.
//   - Only `mtiles` row tiles are active at step s (rows [0,(16-s)*8)).
//   - Waves pair up (adjacent IDs, same block): each computes half of K,
//     kh=1 dumps its 4 gate accumulators to LDS, kh=0 combines + epilogue.
//   g = x_pre + h @ Whh^T ; gate nonlinearities; c in place; h ping-pong;
//   last_h capture for group t == T-1-step.
// ---------------------------------------------------------------------------
__global__ void lstm_step_kernel(const float* __restrict__ xpre,
                                 const float* __restrict__ hin,
                                 const float* __restrict__ Whh,
                                 float* __restrict__ cc,
                                 float* __restrict__ hout,
                                 float* __restrict__ lasth,
                                 int step, int mtiles) {
  __shared__ float red[4][4][8][32];   // [pair][gate][acc elem][lane] = 16KB
  int lw   = threadIdx.x >> 5;         // local wave 0..7
  int lane = threadIdx.x & 31;
  int w    = blockIdx.x * 8 + lw;
  int kh   = w & 1;                    // K-half
  int wt   = w >> 1;                   // tile index
  int mt   = wt % mtiles;
  int n0   = (wt / mtiles) << 4;       // col tile within D
  int m0   = mt << 4;                  // row tile
  int nn   = lane & 15;
  int mr   = lane & 15;
  int koff = (lane >> 4) << 1;
  int hi   = lane >> 4;
  int kbase = kh * (DD / 2);

  v8f acc[4];
  if (kh == 0) {
    #pragma unroll
    for (int g = 0; g < 4; ++g)
      #pragma unroll
      for (int v = 0; v < 8; ++v) {
        int r = m0 + v + (hi << 3);
        acc[g][v] = xpre[(size_t)r * (4 * DD) + g * DD + n0 + nn];
      }
  } else {
    #pragma unroll
    for (int g = 0; g < 4; ++g)
      #pragma unroll
      for (int v = 0; v < 8; ++v) acc[g][v] = 0.f;
  }

  const float* arow = hin + (size_t)(m0 + mr) * DD + kbase + koff;
  const float* w0 = Whh + (size_t)(0 * DD + n0 + nn) * DD + kbase + koff;
  const float* w1 = Whh + (size_t)(1 * DD + n0 + nn) * DD + kbase + koff;
  const float* w2 = Whh + (size_t)(2 * DD + n0 + nn) * DD + kbase + koff;
  const float* w3 = Whh + (size_t)(3 * DD + n0 + nn) * DD + kbase + koff;

  #pragma unroll 2
  for (int k = 0; k < DD / 2; k += 4) {
    v2f a = *(const v2f*)(arow + k);
    acc[0] = __builtin_amdgcn_wmma_f32_16x16x4_f32(false, a, false,
               *(const v2f*)(w0 + k), (short)0, acc[0], false, false);
    acc[1] = __builtin_amdgcn_wmma_f32_16x16x4_f32(false, a, false,
               *(const v2f*)(w1 + k), (short)0, acc[1], false, false);
    acc[2] = __builtin_amdgcn_wmma_f32_16x16x4_f32(false, a, false,
               *(const v2f*)(w2 + k), (short)0, acc[2], false, false);
    acc[3] = __builtin_amdgcn_wmma_f32_16x16x4_f32(false, a, false,
               *(const v2f*)(w3 + k), (short)0, acc[3], false, false);
  }

  int p = lw >> 1;                     // pair index within block
  if (kh == 1) {
    #pragma unroll
    for (int g = 0; g < 4; ++g)
      #pragma unroll
      for (int v = 0; v < 8; ++v) red[p][g][v][lane] = acc[g][v];
  }
  __syncthreads();
  if (kh == 1) return;

  #pragma unroll
  for (int g = 0; g < 4; ++g)
    #pragma unroll
    for (int v = 0; v < 8; ++v) acc[g][v] += red[p][g][v][lane];

  int tgt = TT - 1 - step;
  #pragma unroll
  for (int v = 0; v < 8; ++v) {
    int r = m0 + v + (hi << 3);
    int d = n0 + nn;
    float si = 1.f / (1.f + expf(-acc[0][v]));
    float sf = 1.f / (1.f + expf(-acc[1][v]));
    float gt = tanhf(acc[2][v]);
    float so = 1.f / (1.f + expf(-acc[3][v]));
    float cold = cc[(size_t)r * DD + d];
    float cnew = sf * cold + si * gt;
    float hnew = so * tanhf(cnew);
    cc[(size_t)r * DD + d] = cnew;
    hout[(size_t)r * DD + d] = hnew;
    if ((r >> 3) == tgt) lasth[(size_t)r * DD + d] = hnew;   // r = t*8+b
  }
}

// ---------------------------------------------------------------------------
// LayerNorm (biased variance, eps=1e-5) + ReLU, one block per row.
// ---------------------------------------------------------------------------
__global__ void ln_relu_kernel(const float* __restrict__ in,
                               const float* __restrict__ g,
                               const float* __restrict__ be,
                               float* __restrict__ out) {
  __shared__ float s1[256], s2[256];
  int r = blockIdx.x, t = threadIdx.x;
  const float* row = in + (size_t)r * DD;
  float s = 0.f, sq = 0.f;
  for (int i = t; i < DD; i += 256) { float v = row[i]; s += v; sq += v * v; }
  s1[t] = s; s2[t] = sq;
  __syncthreads();
  for (int off = 128; off > 0; off >>= 1) {
    if (t < off) { s1[t] += s1[t + off]; s2[t] += s2[t + off]; }
    __syncthreads();
  }
  float mean = s1[0] * (1.0f / DD);
  float var = s2[0] * (1.0f / DD) - mean * mean;
  float inv = rsqrtf(var + 1e-5f);
  for (int i = t; i < DD; i += 256) {
    float v = (row[i] - mean) * inv * g[i] + be[i];
    out[(size_t)r * DD + i] = v > 0.f ? v : 0.f;
  }
}

// ---------------------------------------------------------------------------
extern "C" void kernel_launch(void* const* d_in, const int* in_sizes, int n_in,
                              void* d_out, int out_size, void* d_ws, size_t ws_size,
                              hipStream_t stream) {
  const float* x   = (const float*)d_in[0];
  const float* Wn  = (const float*)d_in[1];
  const float* bn  = (const float*)d_in[2];
  const float* Wv  = (const float*)d_in[3];
  const float* bv  = (const float*)d_in[4];
  const float* Wf  = (const float*)d_in[5];
  const float* bf  = (const float*)d_in[6];
  const float* Wih = (const float*)d_in[7];
  const float* Whh = (const float*)d_in[8];
  const float* bih = (const float*)d_in[9];
  const float* bhh = (const float*)d_in[10];
  const float* Wc  = (const float*)d_in[11];
  const float* bc  = (const float*)d_in[12];
  const float* g_a = (const float*)d_in[13];
  const float* be_a= (const float*)d_in[14];
  const float* Wa  = (const float*)d_in[15];
  const float* ba  = (const float*)d_in[16];
  float* out = (float*)d_out;

  float* ws    = (float*)d_ws;
  float* A0    = ws;               // 128x1024  [T,B] action stream
  float* F1    = A0    + 131072;   // 128x1024
  float* F2    = F1    + 131072;   // 128x1024
  float* ENV   = F2    + 131072;   // 128x2048  [e_n | e_v]
  float* XPRE  = ENV   + 262144;   // 128x4096
  float* H0    = XPRE  + 524288;   // 128x1024  h ping
  float* H1    = H0    + 131072;   // 128x1024  h pong
  float* CC    = H1    + 131072;   // 128x1024  c state
  float* LASTH = CC    + 131072;   // 128x1024
  float* UNR   = LASTH + 131072;   // 128x1024
  float* LNR   = UNR   + 131072;   // 128x1024

  // 1) adaptive average pool
  pool_kernel<<<1536, 256, 0, stream>>>(x, A0, F1, F2);
  // 2) e_n / e_v into concatenated buffer
  gemm_kernel<0><<<64, 256, 0, stream>>>(F1, DD, Wn, DD, bn, nullptr,
                                         ENV, 2 * DD, 0,  nullptr, DD, DD);
  gemm_kernel<0><<<64, 256, 0, stream>>>(F2, DD, Wv, DD, bv, nullptr,
                                         ENV, 2 * DD, DD, nullptr, DD, DD);
  // 3) fuse_ev -> h0 (H0), c0 (CC), remapped to [T,B]
  gemm_kernel<1><<<128, 256, 0, stream>>>(ENV, 2 * DD, Wf, 2 * DD, bf, nullptr,
                                          H0, 0, 0, CC, 2 * DD, 2 * DD);
  // 4) x_pre = a @ Wih^T + bih + bhh
  gemm_kernel<0><<<256, 256, 0, stream>>>(A0, DD, Wih, DD, bih, bhh,
                                          XPRE, 4 * DD, 0, nullptr, 4 * DD, DD);
  // 5) 16 sequential LSTM steps (pruned rows, split-K=2, h ping-pong)
  float* hin = H0; float* hout = H1;
  for (int s = 0; s < TT; ++s) {
    int mtiles = (TT - s + 1) >> 1;          // ceil((16-s)/2) active row tiles
    lstm_step_kernel<<<mtiles * 16, 256, 0, stream>>>(XPRE, hin, Whh, CC, hout,
                                                      LASTH, s, mtiles);
    float* tmp = hin; hin = hout; hout = tmp;
  }
  // 6) ActionHead.classifier
  gemm_kernel<0><<<64, 256, 0, stream>>>(LASTH, DD, Wc, DD, bc, nullptr,
                                         UNR, DD, 0, nullptr, DD, DD);
  // 7) LayerNorm + ReLU
  ln_relu_kernel<<<128, 256, 0, stream>>>(UNR, g_a, be_a, LNR);
  // 8) head GEMM (N=1887, ragged) with [T,B]->[B,T] output remap
  gemm_kernel<2><<<118, 256, 0, stream>>>(LNR, DD, Wa, DD, ba, nullptr,
                                          out, NA, 0, nullptr, NA, DD);
}